// ConvSSMParallelScan_13091060318310
// MI455X (gfx1250) — compile-verified
//
#include <hip/hip_runtime.h>

typedef __attribute__((ext_vector_type(2))) float v2f;
typedef __attribute__((ext_vector_type(8))) float v8f;

#define T_STEPS 64
#define FD      64   // circular (zero-padded FFT) domain: 2H = 2W = 64
#define HH      32
#define WW      32

// One workgroup per (batch, channel) plane. 256 threads = 8 waves (wave32).
// State H (64x64 f32, circular domain) and zero-padded x_t both live in LDS.
// Each step:  h_new = Sum_ks P_ks ( H x M_ks^A  +  X x M_ks^B )
// evaluated as a banded 64x192x64 GEMM with V_WMMA_F32_16X16X4_F32 tiles.
// Circulant fragments are precomputed into registers before the t-loop; the
// hot loop is pure ds_load + v_wmma with 4 independent accumulator chains.
__global__ __launch_bounds__(256) void convssm_scan_kernel(
    const float* __restrict__ xg,   // (T, B, C, 32, 32)
    const float* __restrict__ Akg,  // (C, 3, 3)
    const float* __restrict__ Bkg,  // (C, 3, 3)
    float* __restrict__ outg)       // (T, B, C, 32, 32)
{
    __shared__ float Hs[FD][FD + 1];   // state, stride 65 (conflict-free)
    __shared__ float Xs[FD][FD + 1];   // zero-padded x_t, stride 65
    __shared__ float akl[9];
    __shared__ float bkl[9];

    const int plane = blockIdx.x;      // 0..63 (= batch*32 + channel)
    const int chan  = plane & 31;

    const int tid  = threadIdx.x;
    const int wave = tid >> 5;         // wave32
    const int lane = tid & 31;
    const int n    = lane & 15;        // column within tile (B/C/D layout)
    const int kh   = lane >> 4;        // half-select per ISA fragment layout
    const int m    = lane & 15;        // A-frag row within tile

    // Tile assignment: 4x4 grid of 16x16 tiles; wave w owns tiles w and w+8
    // (same column block -> shared B fragments).
    const int i0a = (wave >> 2) * 16;  // 0 or 16
    const int i0b = i0a + 32;          // 32 or 48
    const int j0  = (wave & 3) * 16;   // 0,16,32,48
    const int jn  = j0 + n;

    if (tid < 9) {
        akl[tid] = Akg[chan * 9 + tid];
        bkl[tid] = Bkg[chan * 9 + tid];
    }
    // Zero both planes once. Xs outside [0,32)x[0,32) stays zero forever,
    // which implements the zero-padding branchlessly.
    for (int idx = tid; idx < FD * (FD + 1); idx += 256) {
        ((float*)Hs)[idx] = 0.0f;
        ((float*)Xs)[idx] = 0.0f;
    }
    __syncthreads();

    // ---- Precompute loop-invariant per-lane data (registers) ----
    int raA[3], rbA[3];
#pragma unroll
    for (int ks = 0; ks < 3; ++ks) {
        raA[ks] = (i0a + m - ks) & 63;
        rbA[ks] = (i0b + m - ks) & 63;
    }

    int kkArr[15];
    v2f bfA[15], bfB[15];   // circulant fragments for A-kernel and B-kernel
    {
        int q = 0;
#pragma unroll
        for (int ks = 0; ks < 3; ++ks) {
            const float a0 = akl[ks * 3 + 0], a1 = akl[ks * 3 + 1], a2 = akl[ks * 3 + 2];
            const float b0 = bkl[ks * 3 + 0], b1 = bkl[ks * 3 + 1], b2 = bkl[ks * 3 + 2];
#pragma unroll
            for (int bbk = 0; bbk < 5; ++bbk, ++q) {
                const int kb  = ((j0 >> 2) + bbk + 15) & 15;  // j0/4 - 1 + bbk (mod 16)
                const int kkA = (kb << 2) + 2 * kh;
                kkArr[q] = kkA;
                const int dx = (jn - kkA) & 63;
                const int dy = (jn - kkA - 1) & 63;
                // branchless selects (v_cndmask): in-band circulant taps, else 0
                bfA[q].x = (dx == 0 ? a0 : 0.0f) + (dx == 1 ? a1 : 0.0f) + (dx == 2 ? a2 : 0.0f);
                bfA[q].y = (dy == 0 ? a0 : 0.0f) + (dy == 1 ? a1 : 0.0f) + (dy == 2 ? a2 : 0.0f);
                bfB[q].x = (dx == 0 ? b0 : 0.0f) + (dx == 1 ? b1 : 0.0f) + (dx == 2 ? b2 : 0.0f);
                bfB[q].y = (dy == 0 ? b0 : 0.0f) + (dy == 1 ? b1 : 0.0f) + (dy == 2 ? b2 : 0.0f);
            }
        }
    }

    // ---- Sequential recurrence over T ----
#pragma unroll 1
    for (int t = 0; t < T_STEPS; ++t) {
        const long long xoff = ((long long)t * 64 + plane) * 1024; // in & out layout match

        // stage x_t (32x32) into the zero-padded LDS plane, coalesced
        for (int idx = tid; idx < HH * WW; idx += 256)
            Xs[idx >> 5][idx & 31] = xg[xoff + idx];

        // prefetch next timestep's tile into near caches while we compute
        if (t + 1 < T_STEPS)
            __builtin_prefetch(&xg[xoff + 65536 + (long long)tid * 16], 0, 3);

        __syncthreads();   // Xs ready; previous step's Hs writes visible

        // 4 independent accumulator chains (H-term / X-term, tile0 / tile1)
        // -> 15-deep WMMA dependency chains instead of 30-deep.
        v8f dh0 = {};
        v8f dx0 = {};
        v8f dh1 = {};
        v8f dx1 = {};

        // Banded GEMM over 5 active K-blocks x 3 row shifts (ds_load + v_wmma)
        {
            int q = 0;
#pragma unroll
            for (int ks = 0; ks < 3; ++ks) {
                const int ra = raA[ks];
                const int rb = rbA[ks];
#pragma unroll
                for (int bbk = 0; bbk < 5; ++bbk, ++q) {
                    const int kk = kkArr[q];
                    v2f ha, hb, xa, xb;
                    ha.x = Hs[ra][kk];  ha.y = Hs[ra][kk + 1];
                    hb.x = Hs[rb][kk];  hb.y = Hs[rb][kk + 1];
                    xa.x = Xs[ra][kk];  xa.y = Xs[ra][kk + 1];
                    xb.x = Xs[rb][kk];  xb.y = Xs[rb][kk + 1];
                    dh0 = __builtin_amdgcn_wmma_f32_16x16x4_f32(
                            false, ha, false, bfA[q], (short)0, dh0, false, false);
                    dx0 = __builtin_amdgcn_wmma_f32_16x16x4_f32(
                            false, xa, false, bfB[q], (short)0, dx0, false, false);
                    dh1 = __builtin_amdgcn_wmma_f32_16x16x4_f32(
                            false, hb, false, bfA[q], (short)0, dh1, false, false);
                    dx1 = __builtin_amdgcn_wmma_f32_16x16x4_f32(
                            false, xb, false, bfB[q], (short)0, dx1, false, false);
                }
            }
        }

        // combine partial sums (VALU, overlaps with XDL drain)
        v8f d0, d1;
#pragma unroll
        for (int v = 0; v < 8; ++v) {
            d0[v] = dh0[v] + dx0[v];
            d1[v] = dh1[v] + dx1[v];
        }

        __syncthreads();   // all waves done reading old Hs / Xs

        // Commit new state (D layout: row = i0 + v + 8*kh, col = jn)
#pragma unroll
        for (int v = 0; v < 8; ++v) {
            Hs[i0a + v + 8 * kh][jn] = d0[v];
            Hs[i0b + v + 8 * kh][jn] = d1[v];
        }
        __syncthreads();   // state committed before next step reads it

        // Emit h_t = top-left 32x32 crop (i0a always < 32; gate on j0 only)
        if (j0 < WW) {
#pragma unroll
            for (int v = 0; v < 8; ++v) {
                const int row = i0a + v + 8 * kh;   // < 32 always
                outg[xoff + (long long)row * WW + jn] = d0[v];
            }
        }
    }
}

extern "C" void kernel_launch(void* const* d_in, const int* in_sizes, int n_in,
                              void* d_out, int out_size, void* d_ws, size_t ws_size,
                              hipStream_t stream) {
    (void)in_sizes; (void)n_in; (void)out_size; (void)d_ws; (void)ws_size;
    const float* x  = (const float*)d_in[0];
    const float* Ak = (const float*)d_in[1];
    const float* Bk = (const float*)d_in[2];
    float* out = (float*)d_out;
    // 64 planes (B*C), one workgroup each; 256 threads = 8 wave32 waves.
    convssm_scan_kernel<<<dim3(64), dim3(256), 0, stream>>>(x, Ak, Bk, out);
}